// LossFun_86517821215670
// MI455X (gfx1250) — compile-verified
//
#include <hip/hip_runtime.h>
#include <stdint.h>

#define BB 32
#define PP 24564
#define CC 81
#define GG 24
#define TILE 64                 // priors per TDM tile (64*81*4B = 20736B per DMA)
#define TILES_PER_BLK 8
#define NBLK_X 48               // 48*8*64 = 24576 >= P
#define ROWS_PER_WAVE (TILE / 8)
#define THRESH 0.5f

typedef unsigned int u32;
typedef unsigned long long u64;
typedef __attribute__((ext_vector_type(4))) unsigned int uv4;
typedef __attribute__((ext_vector_type(8))) int iv8;
typedef __attribute__((ext_vector_type(4))) int iv4;

__device__ __forceinline__ float iou_fn(float tx1, float ty1, float tx2, float ty2,
                                        float px1, float py1, float px2, float py2) {
  float iw = fmaxf(fminf(tx2, px2) - fmaxf(tx1, px1), 0.f);
  float ih = fmaxf(fminf(ty2, py2) - fmaxf(ty1, py1), 0.f);
  float inter = iw * ih;
  float at = (tx2 - tx1) * (ty2 - ty1);
  float ap = (px2 - px1) * (py2 - py1);
  return inter / (at + ap - inter);
}

// ---------------- kernel 1: matching ----------------
// per-prior best truth (argmax over g, first occurrence) -> bto/bti;
// per-truth best prior via packed-key atomicMax (LDS partial -> global).
__global__ void k_match(const float4* __restrict__ priors4,
                        const float* __restrict__ truths,
                        float* __restrict__ bto, int* __restrict__ bti,
                        u64* __restrict__ bpikey) {
  int b = blockIdx.y;
  int p = blockIdx.x * blockDim.x + threadIdx.x;
  __shared__ float tr[GG * 4];
  __shared__ u64 sBest[GG];
  if (threadIdx.x < GG * 4) tr[threadIdx.x] = truths[b * GG * 4 + threadIdx.x];
  if (threadIdx.x < GG) sBest[threadIdx.x] = 0ull;
  __syncthreads();
  if (p < PP) {
    float4 pr = priors4[p];               // (cx, cy, w, h)
    float px1 = pr.x - 0.5f * pr.z, py1 = pr.y - 0.5f * pr.w;
    float px2 = pr.x + 0.5f * pr.z, py2 = pr.y + 0.5f * pr.w;
    float best = -1.f; int bg = 0;
    for (int g = 0; g < GG; ++g) {
      float io = iou_fn(tr[g * 4], tr[g * 4 + 1], tr[g * 4 + 2], tr[g * 4 + 3],
                        px1, py1, px2, py2);
      if (io > best) { best = io; bg = g; }   // strict >: first max, like jnp.argmax
      u64 key = ((u64)__float_as_uint(io) << 32) | (u64)(0xFFFFFFFFu - (u32)p);
      atomicMax(&sBest[g], key);              // smaller p wins ties
    }
    bto[(size_t)b * PP + p] = best;
    bti[(size_t)b * PP + p] = bg;
  }
  __syncthreads();
  if (threadIdx.x < GG) atomicMax(&bpikey[b * GG + threadIdx.x], sBest[threadIdx.x]);
}

// ---------------- kernel 2: force each truth's best prior ----------------
__global__ void k_force(const u64* __restrict__ bpikey,
                        float* __restrict__ bto, int* __restrict__ bti) {
  int b = threadIdx.x;
  if (b < BB) {
    for (int g = 0; g < GG; ++g) {  // sequential: duplicate priors -> last wins
      u32 p = 0xFFFFFFFFu - (u32)(bpikey[b * GG + g] & 0xFFFFFFFFull);
      bto[(size_t)b * PP + p] = 2.0f;
      bti[(size_t)b * PP + p] = (int)g;
    }
  }
}

// ---------------- TDM: stage one TILE x 81 f32 tile of conf into LDS ----------------
__device__ __forceinline__ void tdm_load(u32 lds_off, u64 gaddr, int rows) {
  uv4 g0 = { 1u /*count=1*/, lds_off,
             (u32)(gaddr & 0xFFFFFFFFull),
             (u32)((gaddr >> 32) & 0x01FFFFFFull) | (2u << 30) /*type=image*/ };
  iv8 g1 = { 0x20000 /*data_size=4B*/,
             (int)(81u << 16)                 /* tensor_dim0 = 81 (bits 79:48 lo) */,
             (int)(((u32)rows) << 16)         /* tensor_dim1 = rows (bits 111:80 lo) */,
             (int)(81u << 16)                 /* tile_dim0 = 81 */,
             rows                             /* tile_dim1 = rows, tile_dim2 = 0 */,
             81                               /* tensor_dim0_stride */,
             0, 0 };
  iv4 g2 = { 0, 0, 0, 0 };
  iv4 g3 = { 0, 0, 0, 0 };
#if defined(__clang_major__) && (__clang_major__ >= 23)
  iv8 g4 = { 0, 0, 0, 0, 0, 0, 0, 0 };
  __builtin_amdgcn_tensor_load_to_lds(g0, g1, g2, g3, g4, 0);
#else
  __builtin_amdgcn_tensor_load_to_lds(g0, g1, g2, g3, 0);
#endif
}

// ---------------- kernel 3: main streaming pass ----------------
// wave-per-prior logsumexp over C=81 from TDM-staged LDS tiles (double buffered),
// CE, positive bookkeeping, smooth-L1 loc loss, neg-rank scratch.
__global__ void __launch_bounds__(256) k_conf(
    const float* __restrict__ conf, const float* __restrict__ locd,
    const float* __restrict__ priors, const float* __restrict__ truths,
    const int* __restrict__ labels,
    const float* __restrict__ bto, const int* __restrict__ bti,
    float* __restrict__ negrank, int* __restrict__ npos,
    float* __restrict__ lossl, float* __restrict__ lossc) {
  const int b = blockIdx.y;
  const int tid = threadIdx.x;
  const int wave = tid >> 5, lane = tid & 31;
  __shared__ float buf[2][TILE * CC];
  __shared__ float redL[8], redC[8];
  __shared__ int redN[8];

  const int tile0 = blockIdx.x * TILES_PER_BLK;
  const u32 lds0 = (u32)(uintptr_t)(&buf[0][0]);
  const u64 cbase = (u64)(uintptr_t)conf;

  if (tid < 32) {                       // wave 0 drives the TDM pipeline
    int p0 = tile0 * TILE;
    int rows = min(TILE, PP - p0);
    tdm_load(lds0, cbase + ((u64)b * PP + (u64)p0) * (u64)CC * 4ull, rows);
  }

  float accL = 0.f, accC = 0.f; int accN = 0;

  for (int t = 0; t < TILES_PER_BLK; ++t) {
    int p0 = (tile0 + t) * TILE;
    int rows = min(TILE, PP - p0);
    int cur = t & 1;
    if (tid < 32) {
      if (t + 1 < TILES_PER_BLK) {
        int q0 = (tile0 + t + 1) * TILE;
        int rows2 = min(TILE, PP - q0);
        tdm_load(lds0 + (u32)(((t + 1) & 1) * TILE * CC * 4),
                 cbase + ((u64)b * PP + (u64)q0) * (u64)CC * 4ull, rows2);
        __builtin_amdgcn_s_wait_tensorcnt(1);   // tile t landed, t+1 in flight
      } else {
        __builtin_amdgcn_s_wait_tensorcnt(0);
      }
    }
    __syncthreads();                            // tile t visible to all waves

    for (int k = 0; k < ROWS_PER_WAVE; ++k) {
      int row = wave * ROWS_PER_WAVE + k;
      if (row < rows) {
        int p = p0 + row;
        const float* r = &buf[cur][row * CC];
        float x0 = (lane < CC)      ? r[lane]      : -1e30f;
        float x1 = (lane + 32 < CC) ? r[lane + 32] : -1e30f;
        float x2 = (lane + 64 < CC) ? r[lane + 64] : -1e30f;
        float mx = fmaxf(x0, fmaxf(x1, x2));
        for (int i = 16; i > 0; i >>= 1) mx = fmaxf(mx, __shfl_xor(mx, i, 32));
        float s = expf(x0 - mx) + expf(x1 - mx) + expf(x2 - mx);
        for (int i = 16; i > 0; i >>= 1) s += __shfl_xor(s, i, 32);
        float lse = mx + logf(s);

        size_t bp = (size_t)b * PP + p;
        float ov = bto[bp];
        int g = bti[bp];
        int tc = (ov < THRESH) ? 0 : (labels[b * GG + g] + 1);
        int slot = tc >> 5;
        float xs = (slot == 0) ? x0 : ((slot == 1) ? x1 : x2);
        float gold = __shfl(xs, tc & 31, 32);
        float ce = lse - gold;                  // = -log_softmax[gold]

        if (lane == 0) {
          bool pos = (tc != 0);
          negrank[bp] = pos ? -1.0f : ce;       // sentinel excludes positives
          if (pos) {
            accC += ce; accN += 1;
            const float4 pr = *(const float4*)&priors[p * 4];
            const float* tb = &truths[(b * GG + g) * 4];
            float gx = ((tb[0] + tb[2]) * 0.5f - pr.x) / (0.1f * pr.z);
            float gy = ((tb[1] + tb[3]) * 0.5f - pr.y) / (0.1f * pr.w);
            float gw = logf((tb[2] - tb[0]) / pr.z) / 0.2f;
            float gh = logf((tb[3] - tb[1]) / pr.w) / 0.2f;
            float tgt[4] = { gx, gy, gw, gh };
            const float4 lp = *(const float4*)&locd[bp * 4];
            float ld[4] = { lp.x, lp.y, lp.z, lp.w };
            for (int j = 0; j < 4; ++j) {
              float d = ld[j] - tgt[j];
              float ad = fabsf(d);
              accL += (ad < 1.f) ? 0.5f * d * d : ad - 0.5f;
            }
          }
        }
      }
    }
    __syncthreads();    // everyone done with buf[cur] before it is re-filled
  }

  if (lane == 0) { redL[wave] = accL; redC[wave] = accC; redN[wave] = accN; }
  __syncthreads();
  if (tid == 0) {
    float sL = 0.f, sC = 0.f; int sN = 0;
    for (int w = 0; w < 8; ++w) { sL += redL[w]; sC += redC[w]; sN += redN[w]; }
    atomicAdd(lossl, sL);
    atomicAdd(lossc, sC);                       // positive part of conf loss
    if (sN) atomicAdd(&npos[b], sN);
  }
}

// ---------------- kernel 4: per-batch hard-negative top-k ----------------
// 64-bit-key binary search (value desc, index asc tie-break); each thread caches
// its <=24 rank values in registers, so all 64 iterations run out of VGPRs.
#define ELEMS_PER_THREAD ((PP + 1023) / 1024)
__global__ void __launch_bounds__(1024) k_select(
    const float* __restrict__ negrank, const int* __restrict__ npos,
    float* __restrict__ lossc) {
  const int b = blockIdx.x;
  const int tid = threadIdx.x;
  const int lane = tid & 31, wave = tid >> 5;
  __shared__ u32 scnt[32];
  __shared__ float ssum[32];
  __shared__ u32 bc_cnt;
  const float* nr = negrank + (size_t)b * PP;
  int np = npos[b];
  int need = min(3 * np, PP - 1);
  need = min(need, PP - np);            // cannot select more negatives than exist
  if (need <= 0) return;                // uniform per block

  float vals[ELEMS_PER_THREAD];
  u64 keys[ELEMS_PER_THREAD];
  int nv = 0;
  for (int p = tid; p < PP; p += 1024) {
    float v = nr[p];
    vals[nv] = v;
    keys[nv] = (v >= 0.f)
        ? ((((u64)__float_as_uint(v) + 1ull) << 32) | (u64)(0xFFFFFFFFu - (u32)p))
        : 0ull;                          // positives never selected
    nv++;
  }

  u64 lo = 1ull, hi = ~0ull;            // find max T with count(key >= T) >= need
  while (lo < hi) {
    u64 mid = lo + ((hi - lo + 1) >> 1);
    u32 c = 0;
    for (int i = 0; i < nv; ++i) c += (keys[i] >= mid);
    for (int i = 16; i > 0; i >>= 1) c += __shfl_down(c, i, 32);
    if (lane == 0) scnt[wave] = c;
    __syncthreads();
    if (tid == 0) {
      u32 tot = 0;
      for (int w = 0; w < 32; ++w) tot += scnt[w];
      bc_cnt = tot;
    }
    __syncthreads();
    u32 tot = bc_cnt;
    __syncthreads();
    if (tot >= (u32)need) lo = mid; else hi = mid - 1;
  }

  float s = 0.f;
  for (int i = 0; i < nv; ++i)
    if (keys[i] >= lo) s += vals[i];    // exactly `need` distinct keys selected
  for (int i = 16; i > 0; i >>= 1) s += __shfl_down(s, i, 32);
  if (lane == 0) ssum[wave] = s;
  __syncthreads();
  if (tid == 0) {
    float tot = 0.f;
    for (int w = 0; w < 32; ++w) tot += ssum[w];
    atomicAdd(lossc, tot);
  }
}

// ---------------- kernel 5: finalize ----------------
__global__ void k_final(const int* __restrict__ npos, const float* __restrict__ lossl,
                        const float* __restrict__ lossc, float* __restrict__ out) {
  float N = 0.f;
  for (int b = 0; b < BB; ++b) N += (float)npos[b];
  out[0] = lossl[0] / N;
  out[1] = lossc[0] / N;
}

extern "C" void kernel_launch(void* const* d_in, const int* in_sizes, int n_in,
                              void* d_out, int out_size, void* d_ws, size_t ws_size,
                              hipStream_t stream) {
  const float* locd   = (const float*)d_in[0];  // [B,P,4]
  const float* conf   = (const float*)d_in[1];  // [B,P,C]
  const float* priors = (const float*)d_in[2];  // [P,4]
  const float* truths = (const float*)d_in[3];  // [B,G,4]
  const int*   labels = (const int*)d_in[4];    // [B,G]
  float* out = (float*)d_out;

  char* ws = (char*)d_ws;
  const size_t BP = (size_t)BB * PP;
  float* bto     = (float*)(ws);
  int*   bti     = (int*)  (ws + BP * 4);
  float* negrank = (float*)(ws + BP * 8);
  u64*   bpikey  = (u64*)  (ws + BP * 12);
  int*   npos    = (int*)  (ws + BP * 12 + (size_t)BB * GG * 8);
  float* lossl   = (float*)(ws + BP * 12 + (size_t)BB * GG * 8 + (size_t)BB * 4);
  float* lossc   = lossl + 1;

  // zero bpikey + npos + accumulators (one contiguous region)
  hipMemsetAsync(bpikey, 0, (size_t)BB * GG * 8 + (size_t)BB * 4 + 8, stream);

  k_match <<<dim3((PP + 255) / 256, BB), 256, 0, stream>>>((const float4*)priors, truths,
                                                           bto, bti, bpikey);
  k_force <<<1, BB, 0, stream>>>(bpikey, bto, bti);
  k_conf  <<<dim3(NBLK_X, BB), 256, 0, stream>>>(conf, locd, priors, truths, labels,
                                                 bto, bti, negrank, npos, lossl, lossc);
  k_select<<<BB, 1024, 0, stream>>>(negrank, npos, lossc);
  k_final <<<1, 1, 0, stream>>>(npos, lossl, lossc, out);
}